// BatchedSemiAttention_20169166422019
// MI455X (gfx1250) — compile-verified
//
#include <hip/hip_runtime.h>
#include <math.h>

#define BB    2
#define NN    50000
#define EE    800000
#define INP   128
#define KEY   64
#define VAL   128
#define MROWS (BB * NN)   // 100000, divisible by 16
#define LN_EPS 1e-5f

typedef __attribute__((ext_vector_type(2))) float v2f;
typedef __attribute__((ext_vector_type(8))) float v8f;

// ---------------------------------------------------------------------------
// Monotonic float<->uint encoding for atomic float max
// ---------------------------------------------------------------------------
__device__ __forceinline__ unsigned enc_f32(float f) {
    unsigned u = __float_as_uint(f);
    return (u & 0x80000000u) ? ~u : (u | 0x80000000u);
}
__device__ __forceinline__ float dec_f32(unsigned u) {
    return (u & 0x80000000u) ? __uint_as_float(u & 0x7FFFFFFFu)
                             : __uint_as_float(~u);
}

// ---------------------------------------------------------------------------
// 0) reduce k_w over KEY axis, and k_b
// ---------------------------------------------------------------------------
__global__ void bsa_reduce_kw(const float* __restrict__ kw,
                              const float* __restrict__ kb,
                              float* __restrict__ kwsum,
                              float* __restrict__ kbsum) {
    int i = threadIdx.x;           // 0..127
    float s = 0.f;
    for (int j = 0; j < KEY; ++j) s += kw[i * KEY + j];
    kwsum[i] = s;
    if (i == 0) {
        float t = 0.f;
        for (int j = 0; j < KEY; ++j) t += kb[j];
        kbsum[0] = t;
    }
}

// ---------------------------------------------------------------------------
// 1) ksum[b*N+n] = x[row,:] . kwsum + kbsum   (one wave per row)
// ---------------------------------------------------------------------------
__global__ void bsa_ksum(const float* __restrict__ x,
                         const float* __restrict__ kwsum,
                         const float* __restrict__ kbsum,
                         float* __restrict__ ksum) {
    int wid  = (blockIdx.x * blockDim.x + threadIdx.x) >> 5;
    int lane = threadIdx.x & 31;
    if (wid >= MROWS) return;
    const float* xr = x + (size_t)wid * INP;
    float s = 0.f;
    #pragma unroll
    for (int i = 0; i < INP / 32; ++i) {
        int c = lane + i * 32;
        s += xr[c] * kwsum[c];
    }
    #pragma unroll
    for (int off = 16; off; off >>= 1) s += __shfl_xor(s, off);
    if (lane == 0) ksum[wid] = s + kbsum[0];
}

// ---------------------------------------------------------------------------
// WMMA core: one wave computes a 16x128 tile of C = A(16x128) * W(128x128)
// A fragment (16x4 f32): lane<16 holds K=0,1 ; lane>=16 holds K=2,3
// B fragment (4x16 f32): symmetric layout (lane halves hold K-halves)
// ---------------------------------------------------------------------------
__device__ __forceinline__ void bsa_wmma_tile(const float* __restrict__ Arow,
                                              const float* __restrict__ Bs,
                                              v8f acc[8], int lane) {
    const int kh = (lane >> 4) << 1;   // 0 or 2
    const int n  = lane & 15;
    for (int k = 0; k < 128; k += 4) {
        v2f a = *(const v2f*)(Arow + k + kh);
        #pragma unroll
        for (int c = 0; c < 8; ++c) {
            v2f bf;
            bf.x = Bs[(k + kh)     * 128 + (c << 4) + n];
            bf.y = Bs[(k + kh + 1) * 128 + (c << 4) + n];
            acc[c] = __builtin_amdgcn_wmma_f32_16x16x4_f32(
                false, a, false, bf, (short)0, acc[c], false, false);
        }
    }
}

// ---------------------------------------------------------------------------
// 2) values = x @ v_w + v_b   (WMMA, weight staged in LDS)
// ---------------------------------------------------------------------------
__global__ void bsa_values_gemm(const float* __restrict__ A,
                                const float* __restrict__ W,
                                const float* __restrict__ bias,
                                float* __restrict__ out) {
    __shared__ float Bs[128 * 128];
    for (int i = threadIdx.x; i < 128 * 128; i += blockDim.x) Bs[i] = W[i];
    __syncthreads();

    int lane    = threadIdx.x & 31;
    int wv      = threadIdx.x >> 5;
    int tile    = blockIdx.x * 8 + wv;
    int rowbase = tile * 16;
    if (rowbase >= MROWS) return;

    v8f acc[8] = {};
    const float* Arow = A + (size_t)(rowbase + (lane & 15)) * 128;
    bsa_wmma_tile(Arow, Bs, acc, lane);

    int n      = lane & 15;
    int rowoff = (lane >> 4) << 3;   // 0 or 8
    #pragma unroll
    for (int c = 0; c < 8; ++c) {
        float bb = bias[(c << 4) + n];
        #pragma unroll
        for (int r = 0; r < 8; ++r)
            out[(size_t)(rowbase + rowoff + r) * 128 + (c << 4) + n] = acc[c][r] + bb;
    }
}

// ---------------------------------------------------------------------------
// 3) segment max of score over rows (ordered-uint atomicMax)
// ---------------------------------------------------------------------------
__global__ void bsa_edge_max(const float* __restrict__ conn,
                             const int* __restrict__ rows,
                             const int* __restrict__ cols,
                             const float* __restrict__ ksum,
                             unsigned* __restrict__ mu) {
    int tid = blockIdx.x * blockDim.x + threadIdx.x;
    if (tid >= EE * BB) return;
    int e = tid >> 1, b = tid & 1;
    float s = conn[e] * ksum[b * NN + cols[e]];
    atomicMax(&mu[rows[e] * BB + b], enc_f32(s));
}

__global__ void bsa_m_finalize(unsigned* __restrict__ mu) {
    int i = blockIdx.x * blockDim.x + threadIdx.x;
    if (i >= NN * BB) return;
    unsigned u = mu[i];
    float m = (u == 0u) ? 0.0f : dec_f32(u);
    if (!isfinite(m)) m = 0.0f;
    ((float*)mu)[i] = m;
}

// ---------------------------------------------------------------------------
// 4) ex = exp(score - m[row]); denom += ex
// ---------------------------------------------------------------------------
__global__ void bsa_edge_exp(const float* __restrict__ conn,
                             const int* __restrict__ rows,
                             const int* __restrict__ cols,
                             const float* __restrict__ ksum,
                             const float* __restrict__ m,
                             float* __restrict__ ex,
                             float* __restrict__ denom) {
    int tid = blockIdx.x * blockDim.x + threadIdx.x;
    if (tid >= EE * BB) return;
    int e = tid >> 1, b = tid & 1;
    int r = rows[e];
    float s = conn[e] * ksum[b * NN + cols[e]];
    float ev = __expf(s - m[r * BB + b]);
    ex[tid] = ev;
    atomicAdd(&denom[r * BB + b], ev);
}

// ---------------------------------------------------------------------------
// 5) scatter: agg[b, row, :] += w * values[b, col, :]   (1 block per edge)
// ---------------------------------------------------------------------------
__global__ void bsa_scatter(const int* __restrict__ rows,
                            const int* __restrict__ cols,
                            const float* __restrict__ ex,
                            const float* __restrict__ denom,
                            const float* __restrict__ values,
                            float* __restrict__ agg) {
    int e = blockIdx.x;
    int b = threadIdx.x >> 7;       // 0..1
    int c = threadIdx.x & 127;
    int r = rows[e], cl = cols[e];
    float d = denom[r * BB + b];
    float w = ex[e * BB + b] / (d == 0.0f ? 1.0f : d);
    float v = values[((size_t)b * NN + cl) * VAL + c];
    atomicAdd(&agg[((size_t)b * NN + r) * VAL + c], w * v);
}

// ---------------------------------------------------------------------------
// 6) out = LayerNorm(SiLU(agg @ out_w + out_b)) * ln_g + ln_b  (fused WMMA)
// ---------------------------------------------------------------------------
__global__ void bsa_out_gemm_ln(const float* __restrict__ A,
                                const float* __restrict__ W,
                                const float* __restrict__ bias,
                                const float* __restrict__ lng,
                                const float* __restrict__ lnb,
                                float* __restrict__ out) {
    __shared__ float Bs[128 * 128];
    for (int i = threadIdx.x; i < 128 * 128; i += blockDim.x) Bs[i] = W[i];
    __syncthreads();

    int lane    = threadIdx.x & 31;
    int wv      = threadIdx.x >> 5;
    int tile    = blockIdx.x * 8 + wv;
    int rowbase = tile * 16;
    if (rowbase >= MROWS) return;

    v8f acc[8] = {};
    const float* Arow = A + (size_t)(rowbase + (lane & 15)) * 128;
    bsa_wmma_tile(Arow, Bs, acc, lane);

    int n = lane & 15;
    // bias + SiLU, accumulate row statistics (per VGPR slot r = two rows
    // split across lane halves, so a 16-lane xor-reduction is per-row exact)
    float sr[8], sq[8];
    #pragma unroll
    for (int r = 0; r < 8; ++r) { sr[r] = 0.f; sq[r] = 0.f; }
    #pragma unroll
    for (int c = 0; c < 8; ++c) {
        float bb = bias[(c << 4) + n];
        #pragma unroll
        for (int r = 0; r < 8; ++r) {
            float v = acc[c][r] + bb;
            v = v / (1.0f + __expf(-v));    // SiLU
            acc[c][r] = v;
            sr[r] += v;
            sq[r] += v * v;
        }
    }
    #pragma unroll
    for (int r = 0; r < 8; ++r) {
        #pragma unroll
        for (int off = 8; off; off >>= 1) {
            sr[r] += __shfl_xor(sr[r], off);
            sq[r] += __shfl_xor(sq[r], off);
        }
    }
    float mu[8], rstd[8];
    #pragma unroll
    for (int r = 0; r < 8; ++r) {
        mu[r] = sr[r] * (1.0f / VAL);
        float var = sq[r] * (1.0f / VAL) - mu[r] * mu[r];
        rstd[r] = rsqrtf(var + LN_EPS);
    }
    int rowoff = (lane >> 4) << 3;
    #pragma unroll
    for (int c = 0; c < 8; ++c) {
        int col = (c << 4) + n;
        float g = lng[col], b2 = lnb[col];
        #pragma unroll
        for (int r = 0; r < 8; ++r)
            out[(size_t)(rowbase + rowoff + r) * VAL + col] =
                (acc[c][r] - mu[r]) * rstd[r] * g + b2;
    }
}

// ---------------------------------------------------------------------------
extern "C" void kernel_launch(void* const* d_in, const int* in_sizes, int n_in,
                              void* d_out, int out_size, void* d_ws, size_t ws_size,
                              hipStream_t stream) {
    const float* x     = (const float*)d_in[0];
    const float* conn  = (const float*)d_in[1];
    const float* k_w   = (const float*)d_in[2];
    const float* k_b   = (const float*)d_in[3];
    const float* v_w   = (const float*)d_in[4];
    const float* v_b   = (const float*)d_in[5];
    const float* out_w = (const float*)d_in[6];
    const float* out_b = (const float*)d_in[7];
    const float* ln_g  = (const float*)d_in[8];
    const float* ln_b  = (const float*)d_in[9];
    const int*   rows  = (const int*)d_in[10];
    const int*   cols  = (const int*)d_in[11];
    float* out = (float*)d_out;

    // Workspace carve-up (256B aligned regions)
    char* p = (char*)d_ws;
    auto alloc = [&](size_t bytes) {
        void* r = (void*)p;
        p += (bytes + 255) & ~(size_t)255;
        return r;
    };
    float*    kwsum  = (float*)alloc(INP * sizeof(float));
    float*    kbsum  = (float*)alloc(sizeof(float));
    float*    ksum   = (float*)alloc((size_t)MROWS * sizeof(float));
    unsigned* mbuf   = (unsigned*)alloc((size_t)NN * BB * sizeof(float));
    float*    denom  = (float*)alloc((size_t)NN * BB * sizeof(float));
    float*    ex     = (float*)alloc((size_t)EE * BB * sizeof(float));
    float*    values = (float*)alloc((size_t)MROWS * VAL * sizeof(float));
    float*    agg    = (float*)alloc((size_t)MROWS * VAL * sizeof(float));

    // zero-init segment buffers (graph-capturable memset nodes)
    hipMemsetAsync(mbuf, 0, (size_t)NN * BB * sizeof(float), stream);
    hipMemsetAsync(denom, 0, (size_t)NN * BB * sizeof(float), stream);
    hipMemsetAsync(agg, 0, (size_t)MROWS * VAL * sizeof(float), stream);

    // 0) reduce key weights
    bsa_reduce_kw<<<1, 128, 0, stream>>>(k_w, k_b, kwsum, kbsum);

    // 1) values = x @ v_w + v_b  (WMMA)
    int gemm_blocks = (MROWS / 16 + 7) / 8;   // 8 tiles per 256-thread block
    bsa_values_gemm<<<gemm_blocks, 256, 0, stream>>>(x, v_w, v_b, values);

    // 2) ksum (wave per row)
    bsa_ksum<<<(MROWS * 32 + 255) / 256, 256, 0, stream>>>(x, kwsum, kbsum, ksum);

    // 3) segment max
    int edge_blocks = (EE * BB + 255) / 256;
    bsa_edge_max<<<edge_blocks, 256, 0, stream>>>(conn, rows, cols, ksum, mbuf);
    bsa_m_finalize<<<(NN * BB + 255) / 256, 256, 0, stream>>>(mbuf);

    // 4) exp + denom
    bsa_edge_exp<<<edge_blocks, 256, 0, stream>>>(conn, rows, cols, ksum,
                                                  (const float*)mbuf, ex, denom);

    // 5) scatter-aggregate (1 block per edge, both batches)
    bsa_scatter<<<EE, 256, 0, stream>>>(rows, cols, ex, denom, values, agg);

    // 6) out GEMM + SiLU + LayerNorm (WMMA, fused epilogue)
    bsa_out_gemm_ln<<<gemm_blocks, 256, 0, stream>>>(agg, out_w, out_b,
                                                     ln_g, ln_b, out);
}